// RaycastOcc_39934605919045
// MI455X (gfx1250) — compile-verified
//
#include <hip/hip_runtime.h>
#include <stdint.h>

// Problem constants (match reference)
#define B_   8
#define H_   256
#define W_   320
#define DZ_  128
#define DY_  128
#define DX_  256
#define NSTEPS_ 120
#define NVOX (B_ * DZ_ * DY_ * DX_)   // 33,554,432 voxels

// ---------------------------------------------------------------------------
// Pass 1: pack float32 occupancy (0.0/1.0) into a bitfield.
// 128 MB streamed once -> 4 MB bitfield (L2-resident for the ray march).
// wave32: ballot gives exactly one 32-bit mask; lane k <-> bit k since flat
// voxel indices map 1:1 to consecutive lanes (DX=256 is 32-aligned).
// ---------------------------------------------------------------------------
__global__ __launch_bounds__(256) void RaycastOcc_pack_kernel(
    const float* __restrict__ occ, uint32_t* __restrict__ bits) {
  int i = blockIdx.x * 256 + (int)threadIdx.x;
  // Prefetch well ahead of the stream (speculative; guarded in-bounds).
  int pf = i + 64 * 1024;
  if (pf < NVOX) __builtin_prefetch(occ + pf, 0, 0);  // global_prefetch_b8
  // Non-temporal: this data is dead after packing, don't pollute L2.
  float v = __builtin_nontemporal_load(occ + i);
  unsigned mask = __builtin_amdgcn_ballot_w32(v != 0.0f);
  if ((threadIdx.x & 31u) == 0u) bits[i >> 5] = mask;
}

// ---------------------------------------------------------------------------
// Pass 2: ray march. One thread per output pixel. Blocks are aligned so every
// block belongs to one batch image -> view-matrix loads are wave-uniform
// (scalar s_load path). Arithmetic mirrors the reference exactly:
//   d    = 0.1f + 0.02f*i                (float32, per step)
//   cam  = (dirx*d, diry*d, d)
//   p_r  = ((R[r][0]*camx + R[r][1]*camy) + R[r][2]*camz) + t[r]
// with explicit _rn intrinsics to forbid FMA contraction (floor() is
// integer-boundary sensitive).
// ---------------------------------------------------------------------------
template <bool PACKED>
__global__ __launch_bounds__(256) void RaycastOcc_march_kernel(
    const void* __restrict__ vol,
    const float* __restrict__ view_matrix,   // B x 4 x 4
    const float* __restrict__ intrin,        // fx fy cx cy
    float* __restrict__ out) {
  const int blocksPerImg = (H_ * W_) / 256;  // 320
  const int b   = (int)blockIdx.x / blocksPerImg;                 // uniform
  const int pix = ((int)blockIdx.x % blocksPerImg) * 256 + (int)threadIdx.x;
  const int h = pix / W_;
  const int w = pix % W_;

  const float fx = intrin[0], fy = intrin[1], cx = intrin[2], cy = intrin[3];

  const float* __restrict__ M = view_matrix + b * 16;             // uniform
  const float R00 = M[0],  R01 = M[1],  R02 = M[2],  tx = M[3];
  const float R10 = M[4],  R11 = M[5],  R12 = M[6],  ty = M[7];
  const float R20 = M[8],  R21 = M[9],  R22 = M[10], tz = M[11];

  // dir = ((x+0.5-cx)/fx, (y+0.5-cy)/fy, 1)
  const float dirx = (((float)w + 0.5f) - cx) / fx;
  const float diry = (((float)h + 0.5f) - cy) / fy;

  float result = 0.0f;
  #pragma unroll 4
  for (int i = 0; i < NSTEPS_; ++i) {
    const float d    = __fadd_rn(0.1f, __fmul_rn(0.02f, (float)i));
    const float camx = __fmul_rn(dirx, d);
    const float camy = __fmul_rn(diry, d);
    // camz == d (dirz = 1)

    const float px = __fadd_rn(
        __fadd_rn(__fadd_rn(__fmul_rn(R00, camx), __fmul_rn(R01, camy)),
                  __fmul_rn(R02, d)), tx);
    const float py = __fadd_rn(
        __fadd_rn(__fadd_rn(__fmul_rn(R10, camx), __fmul_rn(R11, camy)),
                  __fmul_rn(R12, d)), ty);
    const float pz = __fadd_rn(
        __fadd_rn(__fadd_rn(__fmul_rn(R20, camx), __fmul_rn(R21, camy)),
                  __fmul_rn(R22, d)), tz);

    const int ix = (int)floorf(px);
    const int iy = (int)floorf(py);
    const int iz = (int)floorf(pz);

    if ((unsigned)ix < (unsigned)DX_ &&
        (unsigned)iy < (unsigned)DY_ &&
        (unsigned)iz < (unsigned)DZ_) {
      const int gidx = ((b * DZ_ + iz) * DY_ + iy) * DX_ + ix;
      float val;
      if (PACKED) {
        const uint32_t word = ((const uint32_t*)vol)[gidx >> 5];
        val = (float)((word >> (gidx & 31)) & 1u);
      } else {
        val = ((const float*)vol)[gidx];
      }
      result = fmaxf(result, val);
      // Binary occupancy: max saturates at 1 -> exact early exit.
      if (result != 0.0f) break;
    }
  }

  // One store per pixel; output never re-read by this kernel.
  __builtin_nontemporal_store(result, out + (size_t)b * (H_ * W_) + pix);
}

// ---------------------------------------------------------------------------
// Host launcher
// Inputs (setup_inputs order): occ3d (f32, 33,554,432), view_matrix (f32, 128),
// intrinsic_params (f32, 4). Output: f32, 8*1*256*320 = 655,360.
// ---------------------------------------------------------------------------
extern "C" void kernel_launch(void* const* d_in, const int* in_sizes, int n_in,
                              void* d_out, int out_size, void* d_ws, size_t ws_size,
                              hipStream_t stream) {
  const float* occ  = (const float*)d_in[0];
  const float* vm   = (const float*)d_in[1];
  const float* intr = (const float*)d_in[2];
  float* out = (float*)d_out;

  const size_t bitsBytes = (size_t)NVOX / 8;  // 4 MiB
  const int marchBlocks = B_ * ((H_ * W_) / 256);  // 2560

  if (d_ws && ws_size >= bitsBytes) {
    uint32_t* bits = (uint32_t*)d_ws;
    RaycastOcc_pack_kernel<<<NVOX / 256, 256, 0, stream>>>(occ, bits);
    RaycastOcc_march_kernel<true><<<marchBlocks, 256, 0, stream>>>(
        (const void*)bits, vm, intr, out);
  } else {
    // Fallback: gather floats directly (128 MB still fits the 192 MB L2).
    RaycastOcc_march_kernel<false><<<marchBlocks, 256, 0, stream>>>(
        (const void*)occ, vm, intr, out);
  }
}